// OfflinePrepareLayer_53025666236777
// MI455X (gfx1250) — compile-verified
//
#include <hip/hip_runtime.h>

// OfflinePrepareLayer for MI455X (gfx1250).
// Pure bandwidth-bound prep op: ~190 MB of traffic, ~77 MFLOP -> HBM roofline.
// Strategy: 128-bit coalesced loads/stores everywhere, NT hints on streamed
// data (indices in, features out), regular hints on the 2.4 MB pos gather
// table so it stays cache-resident.

typedef float v4f __attribute__((ext_vector_type(4)));
typedef int   v4i __attribute__((ext_vector_type(4)));

#define ENVD 3
#define HISTD 15
#define NODE_COLS 21          // HIST + 2*ENV
#define INV_RADIUS 20.0f      // 1 / 0.05

__device__ __forceinline__ float clip1(float x) {
    return fminf(1.0f, fmaxf(-1.0f, x));
}

// One thread handles 4 consecutive nodes so every global access is a 16B-aligned b128.
__global__ __launch_bounds__(256) void node_feature_kernel(
    const float* __restrict__ pos,        // (N,3)
    const float* __restrict__ hist_v,     // (N,15)
    const float* __restrict__ boundary,   // (6,)
    float* __restrict__ node_feature,     // (N,21)
    float* __restrict__ current_v,        // (N,3)
    int nquads)                           // N/4
{
    int i = blockIdx.x * blockDim.x + threadIdx.x;
    if (i >= nquads) return;

    // 4 rows of hist_v: 60 floats = 15 x b128 (base = i*240 bytes, aligned)
    float hv[4 * HISTD];
    {
        const v4f* h4 = (const v4f*)hist_v + (size_t)i * 15;
        v4f* dst4 = (v4f*)hv;
#pragma unroll
        for (int k = 0; k < 15; ++k)
            dst4[k] = __builtin_nontemporal_load(h4 + k);  // streamed, read-once
    }

    // 4 rows of pos: 12 floats = 3 x b128
    float pv[4 * ENVD];
    {
        const v4f* p4 = (const v4f*)pos + (size_t)i * 3;
        v4f* dst4 = (v4f*)pv;
#pragma unroll
        for (int k = 0; k < 3; ++k)
            dst4[k] = p4[k];
    }

    // boundary is wave-uniform -> scalar loads
    const float b0 = boundary[0], b1 = boundary[1], b2 = boundary[2];
    const float b3 = boundary[3], b4 = boundary[4], b5 = boundary[5];

    // Assemble 4 rows x 21 cols = 84 floats, then 21 x b128 NT stores
    float ob[4 * NODE_COLS];
#pragma unroll
    for (int n = 0; n < 4; ++n) {
#pragma unroll
        for (int j = 0; j < HISTD; ++j)
            ob[n * NODE_COLS + j] = hv[n * HISTD + j];   // (v-0)/1 == v
        const float px = pv[n * ENVD + 0];
        const float py = pv[n * ENVD + 1];
        const float pz = pv[n * ENVD + 2];
        ob[n * NODE_COLS + 15] = clip1((px - b0) * INV_RADIUS);
        ob[n * NODE_COLS + 16] = clip1((px - b1) * INV_RADIUS);
        ob[n * NODE_COLS + 17] = clip1((py - b2) * INV_RADIUS);
        ob[n * NODE_COLS + 18] = clip1((py - b3) * INV_RADIUS);
        ob[n * NODE_COLS + 19] = clip1((pz - b4) * INV_RADIUS);
        ob[n * NODE_COLS + 20] = clip1((pz - b5) * INV_RADIUS);
    }
    {
        v4f* o4 = (v4f*)node_feature + (size_t)i * 21;   // base = i*336 bytes, aligned
        const v4f* s4 = (const v4f*)ob;
#pragma unroll
        for (int k = 0; k < 21; ++k)
            __builtin_nontemporal_store(s4[k], o4 + k);
    }

    // current_v = hist_v[:, :3] : 12 floats = 3 x b128 NT stores
    {
        float cv[4 * ENVD];
#pragma unroll
        for (int n = 0; n < 4; ++n) {
            cv[n * ENVD + 0] = hv[n * HISTD + 0];
            cv[n * ENVD + 1] = hv[n * HISTD + 1];
            cv[n * ENVD + 2] = hv[n * HISTD + 2];
        }
        v4f* o4 = (v4f*)current_v + (size_t)i * 3;       // base = i*48 bytes, aligned
        const v4f* s4 = (const v4f*)cv;
#pragma unroll
        for (int k = 0; k < 3; ++k)
            __builtin_nontemporal_store(s4[k], o4 + k);
    }
}

// One thread handles 4 consecutive edges: b128 NT index loads, b32 gathers
// from the L2-resident pos table, b128 NT feature stores.
__global__ __launch_bounds__(256) void edge_feature_kernel(
    const float* __restrict__ pos,        // (N,3)
    const int* __restrict__ src,          // (E,)
    const int* __restrict__ dst,          // (E,)
    float* __restrict__ edge_feature,     // (E,4)
    int equads)                           // E/4
{
    int i = blockIdx.x * blockDim.x + threadIdx.x;
    if (i >= equads) return;

    const v4i s4 = __builtin_nontemporal_load((const v4i*)src + i);
    const v4i d4 = __builtin_nontemporal_load((const v4i*)dst + i);
    const int ss[4] = {s4.x, s4.y, s4.z, s4.w};
    const int dd[4] = {d4.x, d4.y, d4.z, d4.w};

    v4f* out4 = (v4f*)edge_feature + (size_t)i * 4;
#pragma unroll
    for (int k = 0; k < 4; ++k) {
        const float* ps = pos + (size_t)ss[k] * ENVD;    // random gather, keep cached
        const float* pd = pos + (size_t)dd[k] * ENVD;
        const float rx = (ps[0] - pd[0]) * INV_RADIUS;
        const float ry = (ps[1] - pd[1]) * INV_RADIUS;
        const float rz = (ps[2] - pd[2]) * INV_RADIUS;
        const float rd = sqrtf(rx * rx + ry * ry + rz * rz);
        v4f o;
        o.x = rx; o.y = ry; o.z = rz; o.w = rd;
        __builtin_nontemporal_store(o, out4 + k);
    }
}

extern "C" void kernel_launch(void* const* d_in, const int* in_sizes, int n_in,
                              void* d_out, int out_size, void* d_ws, size_t ws_size,
                              hipStream_t stream) {
    const float* pos      = (const float*)d_in[0];  // (N,3)
    const float* hist_v   = (const float*)d_in[1];  // (N,15)
    const float* boundary = (const float*)d_in[2];  // (6,)
    const int*   src      = (const int*)d_in[3];    // (E,)
    const int*   dst      = (const int*)d_in[4];    // (E,)

    const int n = in_sizes[0] / ENVD;   // 200000
    const int e = in_sizes[3];          // 6400000

    float* out_node = (float*)d_out;                          // N*21 floats
    float* out_edge = out_node + (size_t)n * NODE_COLS;       // E*4 floats
    float* out_cv   = out_edge + (size_t)e * 4;               // N*3 floats

    const int nquads = n / 4;           // N divisible by 4
    const int equads = e / 4;           // E divisible by 4

    const int TPB = 256;
    node_feature_kernel<<<(nquads + TPB - 1) / TPB, TPB, 0, stream>>>(
        pos, hist_v, boundary, out_node, out_cv, nquads);
    edge_feature_kernel<<<(equads + TPB - 1) / TPB, TPB, 0, stream>>>(
        pos, src, dst, out_edge, equads);
}